// GINNet0_12567074308658
// MI455X (gfx1250) — compile-verified
//
#include <hip/hip_runtime.h>

#define N_NODES 50000
#define N_EDGES 800000
#define F_IN    602
#define H_DIM   32
#define C_OUT   41
#define BN_EPS  1e-5f

typedef float v2f __attribute__((ext_vector_type(2)));
typedef float v8f __attribute__((ext_vector_type(8)));

__device__ __forceinline__ v2f ldv2(const float* p) { return *(const v2f*)p; }

// ---------------------------------------------------------------- utilities
__global__ void zero_kernel(float* __restrict__ p, int n) {
  int i = blockIdx.x * blockDim.x + threadIdx.x;
  if (i < n) p[i] = 0.0f;
}

// src[R x C] -> dst[C x R]
__global__ void transpose_kernel(const float* __restrict__ src,
                                 float* __restrict__ dst, int R, int C) {
  int i = blockIdx.x * blockDim.x + threadIdx.x;
  if (i < R * C) {
    int r = i / C, c = i % C;
    dst[(long)c * R + r] = src[i];
  }
}

// ---------------------------------------------------------------- WMMA GEMM
// D[N x P] = act( Aeff[N x K] @ B[K x P] + bias ),  Bt = B^T (P x K, contiguous K)
// PRE:  Aeff = relu(A1 + A2 + preb[k])  (fused GIN combine + MLP-a bias + relu)
// K, P compile-time: guards fold away except the single ragged-P instantiation.
template<int K, int P, bool PRE, bool RELU>
__global__ __launch_bounds__(256)
void gemm_wmma_f32(const float* __restrict__ A1, const float* __restrict__ A2,
                   const float* __restrict__ preb, const float* __restrict__ Bt,
                   const float* __restrict__ bias, float* __restrict__ D, int N) {
  constexpr int CT = (P + 15) / 16;
  const int wave     = (blockIdx.x * blockDim.x + threadIdx.x) >> 5;  // wave32
  const int numTiles = N >> 4;                  // 50000 = 16 * 3125
  if (wave >= numTiles) return;                 // wave-uniform: EXEC stays full
  const int lane  = threadIdx.x & 31;
  const int half  = lane >> 4;                  // A/B: K {0,1} vs {2,3} per half-wave
  const int l     = lane & 15;
  const int rbase = wave << 4;
  const int h2    = 2 * half;

  const float* a1p = A1 + (long)(rbase + l) * K + h2;
  const float* a2p = PRE ? (A2 + (long)(rbase + l) * K + h2) : nullptr;
  const float* pbp = PRE ? (preb + h2) : nullptr;

  const float* btp[CT];
  float msk[CT];
#pragma unroll
  for (int t = 0; t < CT; ++t) {
    const int cb = t * 16 + l;
    const bool valid = ((P & 15) == 0) || (cb < P);
    btp[t] = Bt + (long)(valid ? cb : 0) * K + h2;  // clamped: always in-bounds
    msk[t] = valid ? 1.0f : 0.0f;
  }

  v8f c[CT];
#pragma unroll
  for (int t = 0; t < CT; ++t)
#pragma unroll
    for (int j = 0; j < 8; ++j) c[t][j] = 0.0f;

  // main loop: unguarded (k0, k0+1 < K for every lane while k+4 <= K)
#pragma unroll 4
  for (int k = 0; k + 4 <= K; k += 4) {
    v2f a = ldv2(a1p + k);
    if (PRE) {
      v2f a2 = ldv2(a2p + k);
      v2f pb = ldv2(pbp + k);
      a.x = fmaxf(a.x + a2.x + pb.x, 0.0f);
      a.y = fmaxf(a.y + a2.y + pb.y, 0.0f);
    }
#pragma unroll
    for (int t = 0; t < CT; ++t) {
      v2f b = ldv2(btp[t] + k);
      if ((P & 15) != 0) { b.x *= msk[t]; b.y *= msk[t]; }
      c[t] = __builtin_amdgcn_wmma_f32_16x16x4_f32(false, a, false, b,
                                                   (short)0, c[t], false, false);
    }
  }

  // ragged-K tail (only instantiated for K % 4 != 0, i.e. K = 602)
  if constexpr ((K & 3) != 0) {
    constexpr int kt = K & ~3;
    const int k0 = kt + h2;                      // element index within row
    const float* a1b = a1p - h2;                 // raw row base
    v2f a;
    a.x = (k0     < K) ? a1b[k0]     : 0.0f;
    a.y = (k0 + 1 < K) ? a1b[k0 + 1] : 0.0f;
    if (PRE) {
      const float* a2b = a2p - h2;
      const float* pbb = pbp - h2;
      float a2x = (k0     < K) ? a2b[k0]     + pbb[k0]     : 0.0f;
      float a2y = (k0 + 1 < K) ? a2b[k0 + 1] + pbb[k0 + 1] : 0.0f;
      a.x = fmaxf(a.x + a2x, 0.0f);
      a.y = fmaxf(a.y + a2y, 0.0f);
    }
#pragma unroll
    for (int t = 0; t < CT; ++t) {
      const float* btb = btp[t] - h2;
      v2f b;
      b.x = (k0     < K) ? btb[k0]     : 0.0f;
      b.y = (k0 + 1 < K) ? btb[k0 + 1] : 0.0f;
      if ((P & 15) != 0) { b.x *= msk[t]; b.y *= msk[t]; }
      c[t] = __builtin_amdgcn_wmma_f32_16x16x4_f32(false, a, false, b,
                                                   (short)0, c[t], false, false);
    }
  }

  // D layout: VGPR j -> row rbase + j + 8*half, col cb
#pragma unroll
  for (int t = 0; t < CT; ++t) {
    const int cb = t * 16 + l;
    if (((P & 15) == 0) || (cb < P)) {
      const float bv = bias ? bias[cb] : 0.0f;
      float* dp = D + (long)(rbase + 8 * half) * P + cb;
#pragma unroll
      for (int j = 0; j < 8; ++j) {
        float v = c[t][j] + bv;
        if (RELU) v = fmaxf(v, 0.0f);
        dp[(long)j * P] = v;
      }
    }
  }
}

// ---------------------------------------------------------- SpMM (32-dim) via atomics
__global__ __launch_bounds__(256)
void spmm_scatter32(const float* __restrict__ y, const int* __restrict__ row,
                    const int* __restrict__ col, float* __restrict__ agg, int E) {
  long t = (long)blockIdx.x * blockDim.x + threadIdx.x;
  int e = (int)(t >> 5);
  int f = (int)(t & 31);
  if (e < E) {
    int cs = col[e];
    int rd = row[e];
    atomicAdd(&agg[(long)rd * H_DIM + f], y[(long)cs * H_DIM + f]);
  }
}

// ---------------------------------------------------------- BatchNorm (training stats)
__global__ __launch_bounds__(256)
void bn_stats(const float* __restrict__ t, float* __restrict__ stats, int N) {
  __shared__ float ssum[256];
  __shared__ float ssq[256];
  const int f = threadIdx.x & 31;
  float s = 0.0f, q = 0.0f;
  for (int r = blockIdx.x * 8 + (threadIdx.x >> 5); r < N; r += gridDim.x * 8) {
    float v = t[(long)r * H_DIM + f];
    s += v; q += v * v;
  }
  ssum[threadIdx.x] = s; ssq[threadIdx.x] = q;
  __syncthreads();
  for (int st = 128; st >= 32; st >>= 1) {
    if (threadIdx.x < st) {
      ssum[threadIdx.x] += ssum[threadIdx.x + st];
      ssq[threadIdx.x]  += ssq[threadIdx.x + st];
    }
    __syncthreads();
  }
  if (threadIdx.x < 32) {
    atomicAdd(&stats[f],      ssum[threadIdx.x]);
    atomicAdd(&stats[32 + f], ssq[threadIdx.x]);
  }
}

__global__ __launch_bounds__(256)
void bn_apply(const float* __restrict__ t, const float* __restrict__ stats,
              const float* __restrict__ g, const float* __restrict__ be,
              float* __restrict__ h, int N) {
  int i = blockIdx.x * blockDim.x + threadIdx.x;
  if (i >= N * H_DIM) return;
  int f = i & 31;
  float invN = 1.0f / (float)N;
  float m = stats[f] * invN;
  float v = stats[32 + f] * invN - m * m;   // biased variance (jnp.var)
  h[i] = (t[i] - m) * rsqrtf(v + BN_EPS) * g[f] + be[f];
}

// ---------------------------------------------------------------- launcher
extern "C" void kernel_launch(void* const* d_in, const int* in_sizes, int n_in,
                              void* d_out, int out_size, void* d_ws, size_t ws_size,
                              hipStream_t stream) {
  const float* x   = (const float*)d_in[0];
  const int*   row = (const int*)d_in[1];
  const int*   col = (const int*)d_in[2];
  const float* W1a = (const float*)d_in[3];
  const float* b1a = (const float*)d_in[4];
  const float* W1b = (const float*)d_in[5];
  const float* b1b = (const float*)d_in[6];
  const float* g1  = (const float*)d_in[7];
  const float* be1 = (const float*)d_in[8];
  const float* W2a = (const float*)d_in[9];
  const float* b2a = (const float*)d_in[10];
  const float* W2b = (const float*)d_in[11];
  const float* b2b = (const float*)d_in[12];
  const float* g2  = (const float*)d_in[13];
  const float* be2 = (const float*)d_in[14];
  const float* Wf1 = (const float*)d_in[15];
  const float* bf1 = (const float*)d_in[16];
  const float* Wf2 = (const float*)d_in[17];
  const float* bf2 = (const float*)d_in[18];
  float* out = (float*)d_out;

  float* ws       = (float*)d_ws;
  const size_t nh = (size_t)N_NODES * H_DIM;          // 1.6M floats
  float* bufY     = ws;                               // GEMM-a output / head tmp
  float* bufAgg   = bufY   + nh;                      // neighbor aggregate
  float* bufT     = bufAgg + nh;                      // pre-BN MLP output
  float* bufH     = bufT   + nh;                      // post-BN hidden
  float* stats    = bufH   + nh;                      // 64 floats: sum / sumsq
  float* wt1a     = stats  + 64;                      // W1a^T : 32 x 602
  float* wt1b     = wt1a   + (size_t)F_IN * H_DIM;    // 32 x 32
  float* wt2a     = wt1b   + H_DIM * H_DIM;
  float* wt2b     = wt2a   + H_DIM * H_DIM;
  float* wtf1     = wt2b   + H_DIM * H_DIM;
  float* wtf2     = wtf1   + H_DIM * H_DIM;           // Wf2^T : 41 x 32

  const int numTiles   = N_NODES / 16;                // 3125
  const int gemmBlocks = (numTiles + 7) / 8;          // 8 waves / 256-thread block
  const int nhi        = (int)nh;
  const int ewBlocks   = (N_EDGES * 32 + 255) / 256;  // 100000
  dim3 tB(256);

  // transpose weights (B^T gives contiguous-K b64 fragment loads in the GEMMs)
  transpose_kernel<<<(F_IN * H_DIM + 255) / 256, tB, 0, stream>>>(W1a, wt1a, F_IN, H_DIM);
  transpose_kernel<<<(H_DIM * H_DIM + 255) / 256, tB, 0, stream>>>(W1b, wt1b, H_DIM, H_DIM);
  transpose_kernel<<<(H_DIM * H_DIM + 255) / 256, tB, 0, stream>>>(W2a, wt2a, H_DIM, H_DIM);
  transpose_kernel<<<(H_DIM * H_DIM + 255) / 256, tB, 0, stream>>>(W2b, wt2b, H_DIM, H_DIM);
  transpose_kernel<<<(H_DIM * H_DIM + 255) / 256, tB, 0, stream>>>(Wf1, wtf1, H_DIM, H_DIM);
  transpose_kernel<<<(H_DIM * C_OUT + 255) / 256, tB, 0, stream>>>(Wf2, wtf2, H_DIM, C_OUT);

  // ---------------- conv1 ----------------
  zero_kernel<<<(nhi + 255) / 256, tB, 0, stream>>>(bufAgg, nhi);
  zero_kernel<<<1, 64, 0, stream>>>(stats, 64);
  // y = x @ W1a   (dominant GEMM: 50000 x 602 x 32, fp32 WMMA)
  gemm_wmma_f32<F_IN, H_DIM, false, false><<<gemmBlocks, tB, 0, stream>>>(
      x, nullptr, nullptr, wt1a, nullptr, bufY, N_NODES);
  // agg = segment_sum(y[col], row)  — SpMM moved into 32-dim space
  spmm_scatter32<<<ewBlocks, tB, 0, stream>>>(bufY, row, col, bufAgg, N_EDGES);
  // t = relu(y + agg + b1a) @ W1b + b1b
  gemm_wmma_f32<H_DIM, H_DIM, true, false><<<gemmBlocks, tB, 0, stream>>>(
      bufY, bufAgg, b1a, wt1b, b1b, bufT, N_NODES);
  bn_stats<<<256, tB, 0, stream>>>(bufT, stats, N_NODES);
  bn_apply<<<(nhi + 255) / 256, tB, 0, stream>>>(bufT, stats, g1, be1, bufH, N_NODES);

  // ---------------- conv2 ----------------
  zero_kernel<<<(nhi + 255) / 256, tB, 0, stream>>>(bufAgg, nhi);
  zero_kernel<<<1, 64, 0, stream>>>(stats, 64);
  gemm_wmma_f32<H_DIM, H_DIM, false, false><<<gemmBlocks, tB, 0, stream>>>(
      bufH, nullptr, nullptr, wt2a, nullptr, bufY, N_NODES);
  spmm_scatter32<<<ewBlocks, tB, 0, stream>>>(bufY, row, col, bufAgg, N_EDGES);
  gemm_wmma_f32<H_DIM, H_DIM, true, false><<<gemmBlocks, tB, 0, stream>>>(
      bufY, bufAgg, b2a, wt2b, b2b, bufT, N_NODES);
  bn_stats<<<256, tB, 0, stream>>>(bufT, stats, N_NODES);
  bn_apply<<<(nhi + 255) / 256, tB, 0, stream>>>(bufT, stats, g2, be2, bufH, N_NODES);

  // ---------------- head ----------------
  // h3 = relu(h2 @ Wf1 + bf1)
  gemm_wmma_f32<H_DIM, H_DIM, false, true><<<gemmBlocks, tB, 0, stream>>>(
      bufH, nullptr, nullptr, wtf1, bf1, bufY, N_NODES);
  // out = h3 @ Wf2 + bf2   (P = 41, masked third tile)
  gemm_wmma_f32<H_DIM, C_OUT, false, false><<<gemmBlocks, tB, 0, stream>>>(
      bufY, nullptr, nullptr, wtf2, bf2, out, N_NODES);
}